// TurboQuantEmbedding_51479478010526
// MI455X (gfx1250) — compile-verified
//
#include <hip/hip_runtime.h>
#include <stdint.h>

// MI455X / gfx1250 (CDNA5, wave32). Bandwidth-bound kernel (~192 MB traffic
// -> ~8.3us floor @ 23.3 TB/s). FWHT-256 done as H16*Y*H16 via
// V_WMMA_F32_16X16X4_F32 (f32 end-to-end, matches f32 reference precision).

typedef __attribute__((ext_vector_type(2))) float v2f;
typedef __attribute__((ext_vector_type(8))) float v8f;

// ---------------------------------------------------------------------------
// Compile-time reproduction of the reference SIGNS:
//   np.random.RandomState(42 + g*256).choice([-1,1], 256)
// Legacy RandomState.choice -> randint(0,2): one 32-bit MT19937 tempered
// output per sample, masked with 0x1 (mask covers range -> no rejection).
// Bit set => index 1 => +1.0 ; clear => -1.0.
// ---------------------------------------------------------------------------
struct SignBits { uint32_t b[4][8]; };

constexpr SignBits make_signbits() {
    SignBits sb{};
    for (int g = 0; g < 4; ++g) {
        uint32_t mt[624] = {};
        mt[0] = 42u + (uint32_t)g * 256u;
        for (int i = 1; i < 624; ++i)
            mt[i] = 1812433253u * (mt[i - 1] ^ (mt[i - 1] >> 30)) + (uint32_t)i;
        // One in-place twist; index order reproduces genrand_int32's block
        // regeneration exactly (wrapped +397 reads updated entries).
        for (int i = 0; i < 624; ++i) {
            uint32_t y = (mt[i] & 0x80000000u) | (mt[(i + 1) % 624] & 0x7fffffffu);
            uint32_t v = mt[(i + 397) % 624] ^ (y >> 1);
            if (y & 1u) v ^= 0x9908b0dfu;
            mt[i] = v;
        }
        for (int k = 0; k < 256; ++k) {
            uint32_t y = mt[k];
            y ^= y >> 11;
            y ^= (y << 7) & 0x9d2c5680u;
            y ^= (y << 15) & 0xefc60000u;
            y ^= y >> 18;
            if (y & 1u) sb.b[g][k >> 5] |= (1u << (k & 31));
        }
    }
    return sb;
}

__constant__ SignBits g_signs = make_signbits();

// Sylvester Hadamard entry: H16[i][j] = (-1)^popcount(i&j)
__device__ __forceinline__ float hval(int i, int j) {
    return (__popc(i & j) & 1) ? -1.0f : 1.0f;
}

// ---------------------------------------------------------------------------
// One wave per token. Per group g (4 groups of 256 elements):
//   stage 1: W  = H16 * Y        (Y decoded into the B slot from LDS)
//   stage 2: Zt = H16 * W^T      ( = (H16 * Y * H16)^T, H symmetric )
//   out[p=16*n+m] = Zt[m][n] * sign[g][p] * norm[g] / 256
// ---------------------------------------------------------------------------
__global__ __launch_bounds__(256) void tq_embed_kernel(
    const int* __restrict__ ids,          // [T] token ids
    const uint32_t* __restrict__ packed,  // [V][512] one byte (2 nibbles) per word
    const float* __restrict__ norms,      // [V][4]
    const float* __restrict__ cb,         // [16]
    float* __restrict__ out,              // [T][1024]
    int n_tok)
{
    __shared__ uint32_t s_pack[8][512];     // 2 KB packed row per wave
    __shared__ float    s_w[8][16 * 17];    // W matrix, stride-17 pad (bank-conflict-free transpose)
    __shared__ float    s_cb[16];

    const int tid  = threadIdx.x;
    const int wave = tid >> 5;
    const int lane = tid & 31;
    const int n    = lane & 15;   // N (and M for the A operand) index
    const int h    = lane >> 4;   // lane half selects K/M sub-block

    if (tid < 16) s_cb[tid] = cb[tid];
    __syncthreads();

    const int tok = blockIdx.x * 8 + wave;
    if (tok >= n_tok) return;     // wave-uniform

    const int id = ids[tok];

    // A-operand chunks of H16 are pure functions of the lane id: identical
    // for both matmul stages and all 4 groups. Compute the 8 values once
    // and keep them resident (8 VGPRs) instead of cmp/cndmask per WMMA.
    v2f ha[4];
#pragma unroll
    for (int k = 0; k < 4; ++k) {
        const int r0 = 4 * k + 2 * h;
        ha[k].x = hval(n, r0);
        ha[k].y = hval(n, r0 + 1);
    }

    // Stage the packed row into LDS: 4x coalesced global_load_b128 per lane.
    {
        const uint4* src = (const uint4*)(packed + (size_t)id * 512u);
        uint4* dst = (uint4*)&s_pack[wave][0];
#pragma unroll
        for (int j = 0; j < 4; ++j)
            dst[j * 32 + lane] = src[j * 32 + lane];
    }
    // CDNA5 split counters: ensure row is resident in LDS before gathers.
    asm volatile("s_wait_loadcnt 0\n\ts_wait_dscnt 0" ::: "memory");

    float* outw = out + (size_t)tok * 1024u;
    float* wm = &s_w[wave][0];

#pragma unroll
    for (int g = 0; g < 4; ++g) {
        const uint32_t* pk = &s_pack[wave][g * 128];

        // ---- stage 1: W = H16 * Y ------------------------------------
        // A (16x4 chunk of H16): lane holds M=n, K = 4k+2h, 4k+2h+1
        // B (4x16 chunk of Y):   lane holds rows 4k+2h, 4k+2h+1 at col n
        //   Y[r][c] = cb[nibble(p = 16r + c)]
        v8f acc = {0.f, 0.f, 0.f, 0.f, 0.f, 0.f, 0.f, 0.f};
#pragma unroll
        for (int k = 0; k < 4; ++k) {
            v2f b;
            const int w0 = 32 * k + 16 * h + (n >> 1);  // word index of p0 = 16*(4k+2h) + n
            const int sh = (n & 1) * 4;                 // nibble select
            b.x = s_cb[(pk[w0]     >> sh) & 15];
            b.y = s_cb[(pk[w0 + 8] >> sh) & 15];        // p0 + 16 -> word + 8
            acc = __builtin_amdgcn_wmma_f32_16x16x4_f32(
                false, ha[k], false, b, (short)0, acc, false, false);
        }

        // D layout: vgpr v, lane = n + 16*h  holds  W[v + 8h][n]
        // Store with row stride 17 so the transposed read hits 16 distinct banks.
#pragma unroll
        for (int v = 0; v < 8; ++v)
            wm[(v + 8 * h) * 17 + n] = acc[v];
        asm volatile("s_wait_dscnt 0" ::: "memory");

        // ---- stage 2: Zt = H16 * W^T ---------------------------------
        // B chunk: b.x = W^T[4k+2h][n] = W[n][4k+2h]
        v8f acc2 = {0.f, 0.f, 0.f, 0.f, 0.f, 0.f, 0.f, 0.f};
#pragma unroll
        for (int k = 0; k < 4; ++k) {
            const int r0 = 4 * k + 2 * h;
            v2f b;
            b.x = wm[n * 17 + r0];
            b.y = wm[n * 17 + r0 + 1];
            acc2 = __builtin_amdgcn_wmma_f32_16x16x4_f32(
                false, ha[k], false, b, (short)0, acc2, false, false);
        }

        // ---- sign * norm/256, store 8 contiguous floats per lane -----
        // Zt[m'][n'] with m' = v + 8h, n' = n  ==>  out p = 16n + 8h + v
        const float ns = norms[(size_t)id * 4u + g] * 0.00390625f; // norm/256
        const int pbase = 16 * n + 8 * h;
        const uint32_t sbyte = (g_signs.b[g][pbase >> 5] >> (pbase & 31)) & 0xFFu;
        float o[8];
#pragma unroll
        for (int v = 0; v < 8; ++v)
            o[v] = acc2[v] * (((sbyte >> v) & 1u) ? ns : -ns);
        float4* dst = (float4*)(outw + g * 256 + pbase);
        dst[0] = make_float4(o[0], o[1], o[2], o[3]);
        dst[1] = make_float4(o[4], o[5], o[6], o[7]);
    }
}

extern "C" void kernel_launch(void* const* d_in, const int* in_sizes, int n_in,
                              void* d_out, int out_size, void* d_ws, size_t ws_size,
                              hipStream_t stream) {
    (void)n_in; (void)out_size; (void)d_ws; (void)ws_size;
    const int*      ids    = (const int*)d_in[0];       // input_ids (8*4096)
    const uint32_t* packed = (const uint32_t*)d_in[1];  // indices_packed (V*512)
    const float*    norms  = (const float*)d_in[2];     // weight_norms (V*4)
    const float*    cb     = (const float*)d_in[3];     // codebook (16)
    float*          out    = (float*)d_out;

    const int n_tok = in_sizes[0];                      // 32768
    const int blocks = (n_tok + 7) / 8;                 // 8 waves (tokens) per block
    hipLaunchKernelGGL(tq_embed_kernel, dim3(blocks), dim3(256), 0, stream,
                       ids, packed, norms, cb, out, n_tok);
}